// FirstLevelAttention_72507637891622
// MI455X (gfx1250) — compile-verified
//
#include <hip/hip_runtime.h>
#include <cstdint>

// Problem shape (fixed by the reference): gather rows of [B, L, D] at [B, NE].
//   sentence_matrix : [64, 4096, 256] f32
//   entity_pos_index: [64, 2] i32
//   out             : [64, 2, 256] f32
// Pure data movement: ~256 KB total -> launch-latency bound on MI455X
// (23.3 TB/s => ~11 ns of traffic). Strategy: one 64-thread block per output
// row (128 blocks), one b128 per lane, routed through the CDNA5 async
// global<->LDS DMA path (ASYNCcnt) so no VGPR data registers are involved.

namespace {
constexpr int kB  = 64;
constexpr int kL  = 4096;
constexpr int kD  = 256;   // floats per row (1024 bytes)
constexpr int kNE = 2;
}

__global__ __launch_bounds__(64) void gather_rows_async_kernel(
    const float* __restrict__ sent,   // [B, L, D]
    const int*   __restrict__ idx,    // [B, NE]
    float*       __restrict__ out)    // [B, NE, D]
{
    __shared__ float row_buf[kD];     // 1024 B staging tile in LDS

    const int row = blockIdx.x;       // 0 .. B*NE-1
    const int b   = row >> 1;         // NE == 2
    const int t   = threadIdx.x;      // 0 .. 63, one float4 per lane

    // Uniform per block -> scalar load (s_load_b32).
    int pos = idx[row];
    // Defensive clamp (indices are generated in [0, L) by the reference).
    pos &= (kL - 1);

    const float* src = sent + ((size_t)b * kL + (size_t)pos) * kD;
    float*       dst = out  + (size_t)row * kD;

    // Byte addresses for this lane's 16-byte slice.
    const uint32_t lds_off = (uint32_t)(uintptr_t)(void*)row_buf + (uint32_t)(t * 16);
    const uint64_t gsrc    = (uint64_t)(uintptr_t)(src + t * 4);
    const uint64_t gdst    = (uint64_t)(uintptr_t)(dst + t * 4);

    // Async DMA: global -> LDS (16 B per lane, fully coalesced 1024-B burst).
    asm volatile("global_load_async_to_lds_b128 %0, %1, off"
                 :
                 : "v"(lds_off), "v"(gsrc)
                 : "memory");

    // Each lane's store consumes only the LDS bytes its own load wrote,
    // so a same-wave ASYNCcnt drain is the only sync required.
    asm volatile("s_wait_asynccnt 0" ::: "memory");

    // Async DMA: LDS -> global.
    asm volatile("global_store_async_from_lds_b128 %0, %1, off"
                 :
                 : "v"(gdst), "v"(lds_off)
                 : "memory");

    // Drain outstanding async stores before wave end (S_ENDPGM also implies
    // wait-idle; explicit for clarity).
    asm volatile("s_wait_asynccnt 0" ::: "memory");
}

extern "C" void kernel_launch(void* const* d_in, const int* in_sizes, int n_in,
                              void* d_out, int out_size, void* d_ws, size_t ws_size,
                              hipStream_t stream) {
    (void)in_sizes; (void)n_in; (void)out_size; (void)d_ws; (void)ws_size;

    const float* sent = (const float*)d_in[0];
    const int*   idx  = (const int*)d_in[1];
    float*       out  = (float*)d_out;

    dim3 grid(kB * kNE);   // 128 single-row workgroups, spread across WGPs
    dim3 block(64);        // 2 waves, one b128 per lane

    hipLaunchKernelGGL(gather_rows_async_kernel, grid, block, 0, stream,
                       sent, idx, out);
}